// MyCell_64647847740110
// MI455X (gfx1250) — compile-verified
//
#include <hip/hip_runtime.h>
#include <stdint.h>

// Ball query on MI455X (gfx1250, wave32).
//
// d2[m][n] = |x_m|^2 + |h_n|^2 - 2 x.h  < r^2
// Rearranged:  (-2 x.h + |h_n|^2)  <  r^2 - |x_m|^2   (per-row threshold)
// LHS computed as 16x16 tiles by V_WMMA_F32_16X16X4_F32 with augmented K=4:
//   A row m = (-2x0, -2x1, -2x2, 1)   B col n = (h0, h1, h2, |h|^2)
//
// f32 16x4 A layout (ISA 7.12.2): lanes 0-15 hold M=lane K={0,1} in
// VGPR{0,1}; lanes 16-31 hold M=lane-16 K={2,3}.  B (4x16) mirrors with
// M<->N.  C/D: VGPR v, lanes 0-15 -> (row v, col lane), lanes 16-31 ->
// (row v+8, col lane-16), so ballot(c[v] < thr) yields a 32-bit hit mask
// whose bit index equals the column offset for two rows at once.
//
// 2 tiles (32 columns) per loop trip: the two 16-bit row masks concatenate
// into one 32-bit mask whose bit index == column offset in [0,32), so the
// ordered first-two extraction and the early-exit ballot amortize 2x.

typedef float v2f __attribute__((ext_vector_type(2)));
typedef float v8f __attribute__((ext_vector_type(8)));

#define R2     0.01f
#define M_TOT  4096
#define N_TOT  16384
#define SENT   N_TOT

__global__ __launch_bounds__(256) void ballquery_wmma_kernel(
    const float* __restrict__ x,   // (8, 3, 4096)
    const float* __restrict__ h,   // (8, 3, 16384)
    int* __restrict__ out)         // (8, 4096, 2)
{
  const int lane = threadIdx.x & 31;
  const int gw   = blockIdx.x * 8 + (threadIdx.x >> 5); // global wave, 2048 total
  const int b    = gw >> 8;                             // batch
  const int m0   = (gw & 255) << 4;                     // first of 16 query rows

  const int hi = lane >> 4;      // which K-half / which C row-half
  const int r  = lane & 15;      // row (A) / column (B,C) within tile

  const float* xb = x + (size_t)b * 3 * M_TOT;
  const float* hb = h + (size_t)b * 3 * N_TOT;

  // ---- A matrix (loaded once): row m = m0 + r, augmented K dim ----
  const int m = m0 + r;
  const float x0 = xb[0 * M_TOT + m];
  const float x1 = xb[1 * M_TOT + m];
  const float x2 = xb[2 * M_TOT + m];
  v2f amat;
  amat.x = hi ? (-2.0f * x2) : (-2.0f * x0);   // K=2 : K=0
  amat.y = hi ? 1.0f         : (-2.0f * x1);   // K=3 : K=1

  // ---- per-row thresholds r^2 - |x|^2 for the 8 rows in this lane's C half
  float thr[8];
#pragma unroll
  for (int v = 0; v < 8; ++v) {
    const int mi = m0 + 8 * hi + v;
    const float a0 = xb[0 * M_TOT + mi];
    const float a1 = xb[1 * M_TOT + mi];
    const float a2 = xb[2 * M_TOT + mi];
    thr[v] = R2 - (a0 * a0 + a1 * a1 + a2 * a2);
  }

  // Lane r (mirrored in lane r+16) owns running state for query row m0+r.
  int idx0 = SENT, idx1 = SENT;

  for (int n0 = 0; n0 < N_TOT; n0 += 32) {
    // ---- B tiles: cols n0+r (tile0) and n0+16+r (tile1) ----
    const int na = n0 + r;
    const float p0 = hb[0 * N_TOT + na];
    const float p1 = hb[1 * N_TOT + na];
    const float p2 = hb[2 * N_TOT + na];
    const float q0 = hb[0 * N_TOT + na + 16];
    const float q1 = hb[1 * N_TOT + na + 16];
    const float q2 = hb[2 * N_TOT + na + 16];

    v2f bmat0, bmat1;
    bmat0.x = hi ? p2 : p0;                                // K=2 : K=0
    bmat0.y = hi ? (p0 * p0 + p1 * p1 + p2 * p2) : p1;     // K=3 : K=1
    bmat1.x = hi ? q2 : q0;
    bmat1.y = hi ? (q0 * q0 + q1 * q1 + q2 * q2) : q1;

    v8f c0 = {0.f, 0.f, 0.f, 0.f, 0.f, 0.f, 0.f, 0.f};
    v8f c1 = {0.f, 0.f, 0.f, 0.f, 0.f, 0.f, 0.f, 0.f};
    c0 = __builtin_amdgcn_wmma_f32_16x16x4_f32(
        false, amat, false, bmat0, (short)0, c0, false, false);
    c1 = __builtin_amdgcn_wmma_f32_16x16x4_f32(
        false, amat, false, bmat1, (short)0, c1, false, false);

    // One ballot per C VGPR: bits[15:0] = row v hits, bits[31:16] = row v+8.
    unsigned bal0[8], bal1[8];
#pragma unroll
    for (int v = 0; v < 8; ++v)
      bal0[v] = __builtin_amdgcn_ballot_w32(c0[v] < thr[v]);
#pragma unroll
    for (int v = 0; v < 8; ++v)
      bal1[v] = __builtin_amdgcn_ballot_w32(c1[v] < thr[v]);

    // Route the ballot half belonging to this lane's owned row (= r):
    //   row r < 8  -> low  half of bal[r] ; row r >= 8 -> high half of bal[r-8]
    unsigned msel0 = bal0[0];
    unsigned msel1 = bal1[0];
#pragma unroll
    for (int v = 1; v < 8; ++v) {
      const bool pick = ((r & 7) == v);
      msel0 = pick ? bal0[v] : msel0;
      msel1 = pick ? bal1[v] : msel1;
    }
    const unsigned mask0 = (r < 8) ? (msel0 & 0xFFFFu) : (msel0 >> 16);
    const unsigned mask1 = (r < 8) ? (msel1 & 0xFFFFu) : (msel1 >> 16);
    // Concatenated 32-column hit mask: bit p -> column n0 + p.
    const unsigned cmask = mask0 | (mask1 << 16);

    // Columns scan in increasing order -> grab first (up to) two hits.
    if (idx1 == SENT && cmask) {
      const int c0i = n0 + __builtin_ctz(cmask);
      const unsigned m2 = cmask & (cmask - 1);
      if (idx0 == SENT) {
        idx0 = c0i;
        if (m2) idx1 = n0 + __builtin_ctz(m2);
      } else {
        idx1 = c0i;
      }
    }

    // Wave-uniform early exit once every row has both samples (EXEC stays
    // all-1s for the WMMAs above; break condition is scalar).
    if (__builtin_amdgcn_ballot_w32(idx1 != SENT) == 0xFFFFFFFFu) break;
  }

  // ---- write results with reference padding semantics ----
  if (lane < 16) {
    const int pad = (idx0 < SENT) ? idx0 : 0;
    const int o0  = (idx0 < SENT) ? idx0 : pad;
    const int o1  = (idx1 < SENT) ? idx1 : pad;
    const size_t base = ((size_t)b * M_TOT + (size_t)(m0 + r)) * 2;
    out[base + 0] = o0;
    out[base + 1] = o1;
  }
}

extern "C" void kernel_launch(void* const* d_in, const int* in_sizes, int n_in,
                              void* d_out, int out_size, void* d_ws, size_t ws_size,
                              hipStream_t stream) {
  (void)in_sizes; (void)n_in; (void)out_size; (void)d_ws; (void)ws_size;
  const float* x = (const float*)d_in[0];   // (8,3,4096)  float32
  const float* h = (const float*)d_in[1];   // (8,3,16384) float32
  int* out = (int*)d_out;                   // (8,4096,2)  int32

  // 2048 waves = 8 batches x 256 row-groups; 8 waves per 256-thread block.
  ballquery_wmma_kernel<<<256, 256, 0, stream>>>(x, h, out);
}